// BktModel_75015898792592
// MI455X (gfx1250) — compile-verified
//
#include <hip/hip_runtime.h>
#include <stdint.h>

typedef float    v2f __attribute__((ext_vector_type(2)));
typedef float    v8f __attribute__((ext_vector_type(8)));
typedef uint32_t v4u __attribute__((ext_vector_type(4)));
typedef uint32_t v8u __attribute__((ext_vector_type(8)));

#define B0      128
#define KSUB    8
#define TT      1024
#define AA      5
#define SS      1024
#define MAXLEN  8192
#define NPROB   1000
#define EPSV    1e-12f
#define NT      (MAXLEN / 256)   // 32 tiles of 256 timesteps per student

__device__ __forceinline__ float sigmoidf(float x) {
    return 1.0f / (1.0f + __expf(-x));
}

// ---------------------------------------------------------------------------
// Tensor Data Mover: dense 1-D copy of `nelem` f32 from global to LDS.
// D# per CDNA5 ISA (2-group form, VADDR2/3 NULL):
//   group0: count=1 | lds_addr | global_addr[56:0] | type=2
//   group1: data_size=4B, tensor_dim0=tile_dim0=nelem, dim0_stride=nelem,
//           higher dims unused, no multicast/iterate/pad/gather.
// Issue from a single wave (TDM ignores EXEC; one issue per executing wave),
// completion tracked with TENSORcnt.
// ---------------------------------------------------------------------------
__device__ __forceinline__ void tdm_load_f32_to_lds(const void* gsrc,
                                                    uint32_t lds_addr,
                                                    uint32_t nelem) {
    const uint64_t ga = (uint64_t)(uintptr_t)gsrc;
    v4u g0;
    g0[0] = 1u;                                                // count=1, user D#
    g0[1] = lds_addr;                                          // LDS byte address
    g0[2] = (uint32_t)(ga & 0xFFFFFFFFu);                      // global_addr[31:0]
    g0[3] = (uint32_t)((ga >> 32) & 0x1FFFFFFu) | (2u << 30);  // addr[56:32]|type=2
    v8u g1;
    g1[0] = (2u << 16);                        // workgroup_mask=0, data_size=4B
    g1[1] = (nelem & 0xFFFFu) << 16;           // tensor_dim0[15:0]
    g1[2] = (nelem >> 16) | (1u << 16);        // tensor_dim0[31:16], tensor_dim1=1
    g1[3] = (nelem & 0xFFFFu) << 16;           // tile_dim0 = nelem
    g1[4] = 0u;                                // tile_dim1/2 unused
    g1[5] = nelem;                             // tensor_dim0_stride[31:0]
    g1[6] = 0u;                                // stride hi / dim1_stride lo
    g1[7] = 0u;
    asm volatile("tensor_load_to_lds %0, %1" :: "s"(g0), "s"(g1) : "memory");
}

// ---------------------------------------------------------------------------
// Kernel 1: BKT forward filter. One thread per (ability, subsequence) row.
// Problem table staged in LDS by the TDM; int streams read as int4;
// speculative global_prefetch_b8 ahead of the stream.
// result layout: [(b*MAXLEN + l)][a][c] ; trial_id == b*MAXLEN + l directly.
// Padding positions are never written; kernel 2c emits exact zeros there.
// ---------------------------------------------------------------------------
__global__ void bkt_filter_kernel(
    const float* __restrict__ dyn_logits,   // (50,3)
    const float* __restrict__ obs_kc,       // (50,2)
    const float* __restrict__ obs_prob,     // (1000,2)
    const float* __restrict__ abil,         // (5,)
    const int*   __restrict__ trial,        // (S,T)
    const int*   __restrict__ problem,      // (S,T)
    const int*   __restrict__ correct,      // (S,T)
    const int*   __restrict__ kc,           // (S,)
    float*       __restrict__ result)       // (B0, MAXLEN, A, 2)
{
    __shared__ float sprob[NPROB * 2];
    if (threadIdx.x < 32) {   // wave 0 only: single TDM issue per block
        tdm_load_f32_to_lds(obs_prob, (uint32_t)(uintptr_t)sprob, NPROB * 2);
        __builtin_amdgcn_s_wait_tensorcnt(0);
    }
    __syncthreads();

    const int gid = blockIdx.x * blockDim.x + threadIdx.x;
    if (gid >= AA * SS) return;
    const int a = gid / SS;
    const int s = gid - a * SS;

    const int   kcs = kc[s];
    const float pL  = sigmoidf(dyn_logits[kcs * 3 + 0]);
    const float pF  = sigmoidf(dyn_logits[kcs * 3 + 1]);
    const float pI  = sigmoidf(dyn_logits[kcs * 3 + 2]);
    const float oG  = obs_kc[kcs * 2 + 0];
    const float oS  = obs_kc[kcs * 2 + 1];
    const float ab  = abil[a];

    float a0 = 1.0f - pI;
    float a1 = pI;

    const int4* __restrict__ tr4 = (const int4*)(trial   + s * TT);
    const int4* __restrict__ pr4 = (const int4*)(problem + s * TT);
    const int4* __restrict__ co4 = (const int4*)(correct + s * TT);

    bool done = false;
    for (int t4 = 0; t4 < TT / 4 && !done; ++t4) {
        __builtin_prefetch(pr4 + ((t4 + 24) & (TT / 4 - 1)), 0, 0);
        const int4 tr = tr4[t4];
        const int4 pr = pr4[t4];
        const int4 co = co4[t4];
        const int tids[4] = {tr.x, tr.y, tr.z, tr.w};
        const int ps[4]   = {pr.x, pr.y, pr.z, pr.w};
        const int ys[4]   = {co.x, co.y, co.z, co.w};
#pragma unroll
        for (int u = 0; u < 4; ++u) {
            const int tid = tids[u];
            if (tid < 0) { done = true; break; }
            const int p = ps[u];
            const int y = ys[u];

            const float pG  = sigmoidf(oG + sprob[p * 2 + 0] + ab);
            const float pS  = sigmoidf(oS + sprob[p * 2 + 1] - ab);
            const float pc0 = pG;
            const float pc1 = 1.0f - pS;

            const float pcorr = a0 * pc0 + a1 * pc1;
            const int oix = (tid * AA + a) * 2;
            result[oix + 0] = __logf(fmaxf(1.0f - pcorr, EPSV));
            result[oix + 1] = __logf(fmaxf(pcorr, EPSV));

            const float l0 = y ? pc0 : (1.0f - pc0);
            const float l1 = y ? pc1 : (1.0f - pc1);
            float q0 = a0 * l0;
            float q1 = a1 * l1;
            const float rzn = 1.0f / fmaxf(q0 + q1, EPSV);
            q0 *= rzn;  q1 *= rzn;
            a0 = q0 * (1.0f - pL) + q1 * pF;
            a1 = q0 * pL + q1 * (1.0f - pF);
        }
    }
}

// ---------------------------------------------------------------------------
// Kernel 2a: per-(student, tile) partial sums of obs_ll for all 5 abilities.
// One wave per (b,tile); butterfly __shfl_xor reduction.
// ---------------------------------------------------------------------------
__global__ void bkt_tilesum_kernel(
    const float* __restrict__ result,   // (B0, MAXLEN, A, 2)
    const int*   __restrict__ ytrue,    // (B0, MAXLEN)
    float*       __restrict__ tilesum)  // (B0, A, NT)
{
    const int blk  = blockIdx.x;
    const int b    = blk / NT;
    const int tile = blk - b * NT;
    const int lane = threadIdx.x;
    const int ybase = b * MAXLEN;
    const int base  = tile * 256;

    const float2* __restrict__ rr = (const float2*)result;

    float part[AA];
#pragma unroll
    for (int a = 0; a < AA; ++a) part[a] = 0.0f;

    for (int ll = lane; ll < 256; ll += 32) {
        const int l = base + ll;
        const int y = ytrue[ybase + l];
        if (y > -1) {
#pragma unroll
            for (int a = 0; a < AA; ++a) {
                const float2 pr = rr[(ybase + l) * AA + a];
                part[a] += y ? pr.y : pr.x;
            }
        }
    }
#pragma unroll
    for (int a = 0; a < AA; ++a) {
#pragma unroll
        for (int d = 16; d >= 1; d >>= 1)
            part[a] += __shfl_xor(part[a], d, 32);
    }
    if (lane == 0) {
#pragma unroll
        for (int a = 0; a < AA; ++a)
            tilesum[(b * AA + a) * NT + tile] = part[a];
    }
}

// ---------------------------------------------------------------------------
// Kernel 2b: exclusive scan of the 32 tile sums per (student, ability).
// ---------------------------------------------------------------------------
__global__ void bkt_tilescan_kernel(
    const float* __restrict__ tilesum,    // (B0, A, NT)
    float*       __restrict__ tilecarry)  // (B0, A, NT)
{
    const int gid = blockIdx.x * blockDim.x + threadIdx.x;
    if (gid >= B0 * AA) return;
    float carry = 0.0f;
#pragma unroll
    for (int t = 0; t < NT; ++t) {
        tilecarry[gid * NT + t] = carry;
        carry += tilesum[gid * NT + t];
    }
}

// ---------------------------------------------------------------------------
// Kernel 2c: one wave per (student, tile). Exclusive prefix of obs_ll within
// the 256-step tile via an 8-deep V_WMMA_F32_16X16X4_F32 chain per ability:
//   D  = Lstrict(16x16) * X        (intra-chunk exclusive cumsum, 4 WMMAs)
//      + ones(16x16)   * B2        (cross-chunk offsets,          4 WMMAs)
// with X[t][j] = obs[base + 16j + t], B2[r][j] = (r<j) ? T_r : 0,
// T_j = Dstrict[15][j] + X[15][j]. pref = tilecarry + D.
// Triangle predicates are precomputed 0/1 floats; totsh is read with plain
// ds_load_b64 and masked by multiply (no exec-masked loads).
// ---------------------------------------------------------------------------
__global__ void bkt_combine_kernel(
    const float* __restrict__ result,     // (B0, MAXLEN, A, 2)
    const int*   __restrict__ ytrue,      // (B0, MAXLEN)
    const float* __restrict__ tilecarry,  // (B0, A, NT)
    float*       __restrict__ out)        // (B0, MAXLEN, 2)
{
    const int blk  = blockIdx.x;
    const int b    = blk / NT;
    const int tile = blk - b * NT;
    const int lane = threadIdx.x;          // full wave, EXEC all ones at WMMAs
    const int col  = lane & 15;
    const int hi   = lane >> 4;

    const int ybase = b * MAXLEN;
    const int base  = tile * 256;

    const float2* __restrict__ rr = (const float2*)result;

    __shared__ float obs[AA][256];
    __shared__ float totsh[16];

    // Stage obs_ll for this tile.
    for (int ll = lane; ll < 256; ll += 32) {
        const int l = base + ll;
        const int y = ytrue[ybase + l];
        if (y > -1) {
#pragma unroll
            for (int a = 0; a < AA; ++a) {
                const float2 pr = rr[(ybase + l) * AA + a];
                obs[a][ll] = y ? pr.y : pr.x;
            }
        } else {
#pragma unroll
            for (int a = 0; a < AA; ++a) obs[a][ll] = 0.0f;
        }
    }
    __syncthreads();

    // Triangle masks (invariant across abilities): (4k+2*hi+u < col).
    float trimask[8];
#pragma unroll
    for (int k = 0; k < 4; ++k) {
        const int k0 = 4 * k + 2 * hi;
        trimask[2 * k + 0] = (k0     < col) ? 1.0f : 0.0f;
        trimask[2 * k + 1] = (k0 + 1 < col) ? 1.0f : 0.0f;
    }

    const float2* __restrict__ tot2 = (const float2*)totsh;

    float pref[AA][8];
#pragma unroll
    for (int a = 0; a < AA; ++a) {
        const float2* __restrict__ ob2 = (const float2*)&obs[a][0];
        v8f d = {};
        // Pass 1: intra-chunk exclusive cumsum  D += Lstrict * X
#pragma unroll
        for (int k = 0; k < 4; ++k) {
            const float2 xp = ob2[col * 8 + 2 * k + hi];   // X[4k+2hi..+1][col]
            v2f av, bv;
            av.x = trimask[2 * k + 0];
            av.y = trimask[2 * k + 1];
            bv.x = xp.x;
            bv.y = xp.y;
            d = __builtin_amdgcn_wmma_f32_16x16x4_f32(
                    false, av, false, bv, (short)0, d, false, false);
        }
        // Chunk totals T_j = Dstrict[15][j] + X[15][j] (row 15 = d[7], hi lanes)
        if (hi) totsh[col] = d[7] + obs[a][col * 16 + 15];
        __syncthreads();

        // Pass 2: cross-chunk offsets  D += ones * B2, B2[r][j] = (r<j) ? T_r : 0
#pragma unroll
        for (int k = 0; k < 4; ++k) {
            const float2 tp = tot2[2 * k + hi];            // T[4k+2hi..+1]
            v2f av, bv;
            av.x = 1.0f;
            av.y = 1.0f;
            bv.x = trimask[2 * k + 0] * tp.x;
            bv.y = trimask[2 * k + 1] * tp.y;
            d = __builtin_amdgcn_wmma_f32_16x16x4_f32(
                    false, av, false, bv, (short)0, d, false, false);
        }
        const float tc = tilecarry[(b * AA + a) * NT + tile];
#pragma unroll
        for (int v = 0; v < 8; ++v) pref[a][v] = tc + d[v];
        __syncthreads();   // totsh reused by next ability
    }

    // Emit logpred for the 8 timesteps this lane owns: l = base + 16*col + v + 8*hi
#pragma unroll
    for (int v = 0; v < 8; ++v) {
        const int t = v + 8 * hi;
        const int l = base + col * 16 + t;
        const int y = ytrue[ybase + l];
        float o0 = 0.0f, o1 = 0.0f;
        if (y > -1) {
            float m = -1e30f;
#pragma unroll
            for (int a = 0; a < AA; ++a) m = fmaxf(m, pref[a][v]);
            float ssum = 0.0f;
#pragma unroll
            for (int a = 0; a < AA; ++a) ssum += __expf(pref[a][v] - m);
            const float lse = m + __logf(ssum);

            float r0[AA], r1[AA];
            float m0 = -1e30f, m1 = -1e30f;
#pragma unroll
            for (int a = 0; a < AA; ++a) {
                const float  w  = pref[a][v] - lse;
                const float2 pr = rr[(ybase + l) * AA + a];
                r0[a] = pr.x + w;
                r1[a] = pr.y + w;
                m0 = fmaxf(m0, r0[a]);
                m1 = fmaxf(m1, r1[a]);
            }
            float s0 = 0.0f, s1 = 0.0f;
#pragma unroll
            for (int a = 0; a < AA; ++a) {
                s0 += __expf(r0[a] - m0);
                s1 += __expf(r1[a] - m1);
            }
            o0 = m0 + __logf(s0);
            o1 = m1 + __logf(s1);
        }
        out[(ybase + l) * 2 + 0] = o0;
        out[(ybase + l) * 2 + 1] = o1;
    }
}

// ---------------------------------------------------------------------------
extern "C" void kernel_launch(void* const* d_in, const int* in_sizes, int n_in,
                              void* d_out, int out_size, void* d_ws, size_t ws_size,
                              hipStream_t stream) {
    const float* dyn     = (const float*)d_in[0];
    const float* okc     = (const float*)d_in[1];
    const float* opr     = (const float*)d_in[2];
    const float* abil    = (const float*)d_in[3];
    const int*   trial   = (const int*)d_in[4];
    const int*   problem = (const int*)d_in[5];
    const int*   correct = (const int*)d_in[6];
    const int*   kcv     = (const int*)d_in[7];
    const int*   ytrue   = (const int*)d_in[8];

    float* out = (float*)d_out;                      // (128, 8192, 2)

    const size_t RESBYTES = (size_t)B0 * MAXLEN * AA * 2 * sizeof(float);  // 40 MB
    const size_t TSBYTES  = (size_t)B0 * AA * NT * sizeof(float);          // 80 KB
    float* result    = (float*)d_ws;
    float* tilesum   = (float*)((char*)d_ws + RESBYTES);
    float* tilecarry = (float*)((char*)d_ws + RESBYTES + TSBYTES);

    bkt_filter_kernel<<<(AA * SS + 255) / 256, 256, 0, stream>>>(
        dyn, okc, opr, abil, trial, problem, correct, kcv, result);

    bkt_tilesum_kernel<<<B0 * NT, 32, 0, stream>>>(result, ytrue, tilesum);

    bkt_tilescan_kernel<<<(B0 * AA + 127) / 128, 128, 0, stream>>>(tilesum, tilecarry);

    bkt_combine_kernel<<<B0 * NT, 32, 0, stream>>>(result, ytrue, tilecarry, out);
}